// EnhancedGATModel_13477607375486
// MI455X (gfx1250) — compile-verified
//
#include <hip/hip_runtime.h>
#include <hip/hip_bf16.h>
#include <math.h>

typedef float v2f __attribute__((ext_vector_type(2)));
typedef float v8f __attribute__((ext_vector_type(8)));

// ---------------------------------------------------------------------------
// atomics helpers
// ---------------------------------------------------------------------------
static __device__ __forceinline__ void atomic_add_f32(float* p, float v) {
  unsafeAtomicAdd(p, v);   // global_atomic_add_f32, L2-resident
}

static __device__ __forceinline__ void atomic_max_f32(float* addr, float val) {
  // sign-split trick: positives monotone as int, negatives as uint (reversed)
  if (val >= 0.0f) {
    atomicMax((int*)addr, __float_as_int(val));
  } else {
    atomicMin((unsigned int*)addr, __float_as_uint(val));
  }
}

static __device__ __forceinline__ int gat_src_of(const int* __restrict__ src, int E, int e) {
  return (e < E) ? src[e] : (e - E);   // synthesized self loop
}
static __device__ __forceinline__ int gat_dst_of(const int* __restrict__ dst, int E, int e) {
  return (e < E) ? dst[e] : (e - E);
}

// ---------------------------------------------------------------------------
// weight transpose  Wt[N][K] = W[K][N]  (tiny: <= 32K elements, cache-hot)
// ---------------------------------------------------------------------------
__global__ void gat_transpose(const float* __restrict__ W, float* __restrict__ Wt,
                              int K, int N) {
  int i = blockIdx.x * blockDim.x + threadIdx.x;
  if (i >= K * N) return;
  int k = i / N;
  int n = i - k * N;
  Wt[(size_t)n * K + k] = W[i];
}

// ---------------------------------------------------------------------------
// GEMM:  C[M,N] = A[M,K] @ W[K,N]   via V_WMMA_F32_16X16X4_F32 (wave32)
// One block = 16 rows of A staged in LDS. Each wave accumulates up to two
// 16-wide column tiles per pass so every A fragment feeds two v_wmma issues.
// B comes from the transposed weights Wt[N][K]: the (k, k+1) pair a lane
// needs is contiguous -> single b64 load with immediate offset.
// Fragment layouts per CDNA5 ISA 7.12.2:
//   A 16x4 : lanes 0-15 -> M=0..15 ; VGPR0 K = 2*half, VGPR1 K = 2*half+1
//   B 4x16 : lanes 0-15 -> tile col ; VGPR0 K = 2*half, VGPR1 K = 2*half+1
//   D      : VGPR r -> M = r + 8*half, N = lane&15
// ---------------------------------------------------------------------------
template <int K>
__global__ void gat_gemm_wmma(const float* __restrict__ A,
                              const float* __restrict__ Wt,  // [N][K]
                              float* __restrict__ C,
                              int M, int N) {
  __shared__ float As[16 * K];
  const int m0   = blockIdx.x * 16;
  const int tid  = threadIdx.x;
  const int nthr = blockDim.x;

  for (int i = tid; i < 16 * K; i += nthr) {
    int r  = i / K;
    int k  = i - r * K;
    int gr = m0 + r;
    As[i] = (gr < M) ? A[(size_t)gr * K + k] : 0.0f;
  }
  __syncthreads();

  const int wave   = tid >> 5;
  const int lane   = tid & 31;
  const int half   = lane >> 4;
  const int l16    = lane & 15;
  const int nWaves = nthr >> 5;
  const int nTiles = N >> 4;
  const bool fullM = (m0 + 16 <= M);           // wave-uniform

  const float* __restrict__ ap0 = As + l16 * K + 2 * half;

  for (int ct0 = wave; ct0 < nTiles; ct0 += 2 * nWaves) {
    const int  ct1  = ct0 + nWaves;
    const bool has2 = (ct1 < nTiles);          // wave-uniform
    const int  col0 = ct0 * 16 + l16;
    const int  col1 = has2 ? ct1 * 16 + l16 : col0;

    const float* __restrict__ bp0 = Wt + (size_t)col0 * K + 2 * half;
    const float* __restrict__ bp1 = Wt + (size_t)col1 * K + 2 * half;

    v8f acc0 = {0.f, 0.f, 0.f, 0.f, 0.f, 0.f, 0.f, 0.f};
    v8f acc1 = {0.f, 0.f, 0.f, 0.f, 0.f, 0.f, 0.f, 0.f};

    if (has2) {
#pragma unroll 4
      for (int kk = 0; kk < K; kk += 4) {
        v2f a  = *(const v2f*)(ap0 + kk);      // ds_load_b64, imm offset
        v2f b0 = *(const v2f*)(bp0 + kk);      // global_load_b64, imm offset
        v2f b1 = *(const v2f*)(bp1 + kk);
        acc0 = __builtin_amdgcn_wmma_f32_16x16x4_f32(false, a, false, b0,
                                                     (short)0, acc0, false, false);
        acc1 = __builtin_amdgcn_wmma_f32_16x16x4_f32(false, a, false, b1,
                                                     (short)0, acc1, false, false);
      }
    } else {
#pragma unroll 4
      for (int kk = 0; kk < K; kk += 4) {
        v2f a  = *(const v2f*)(ap0 + kk);
        v2f b0 = *(const v2f*)(bp0 + kk);
        acc0 = __builtin_amdgcn_wmma_f32_16x16x4_f32(false, a, false, b0,
                                                     (short)0, acc0, false, false);
      }
    }

    if (fullM) {                               // uniform fast path: no exec churn
#pragma unroll
      for (int r = 0; r < 8; ++r) {
        size_t row = (size_t)(m0 + r + 8 * half);
        C[row * N + col0] = acc0[r];
        if (has2) C[row * N + col1] = acc1[r];
      }
    } else {
#pragma unroll
      for (int r = 0; r < 8; ++r) {
        int gr = m0 + r + 8 * half;
        if (gr < M) {
          C[(size_t)gr * N + col0] = acc0[r];
          if (has2) C[(size_t)gr * N + col1] = acc1[r];
        }
      }
    }
  }
}

// ---------------------------------------------------------------------------
// per-(node,head) attention logits: al = <h[n,head,:], a[head,:]>
// ---------------------------------------------------------------------------
template <int H, int C>
__global__ void gat_node_logits(const float* __restrict__ hfeat,
                                const float* __restrict__ a_src,
                                const float* __restrict__ a_dst,
                                float* __restrict__ alS,
                                float* __restrict__ alD, int Nn) {
  int t = blockIdx.x * blockDim.x + threadIdx.x;
  if (t >= Nn * H) return;
  int n  = t / H;
  int hd = t - n * H;
  const float* __restrict__ row = hfeat + (size_t)n * (H * C) + hd * C;
  float ss = 0.f, sd = 0.f;
#pragma unroll 4
  for (int c = 0; c < C; ++c) {
    float v = row[c];
    ss = fmaf(v, a_src[hd * C + c], ss);
    sd = fmaf(v, a_dst[hd * C + c], sd);
  }
  alS[t] = ss;
  alD[t] = sd;
}

// ---------------------------------------------------------------------------
// segment max of leaky-relu edge scores into m[dst,head]
// ---------------------------------------------------------------------------
template <int H>
__global__ void gat_edge_max(const int* __restrict__ src, const int* __restrict__ dst,
                             int E, int Etot,
                             const float* __restrict__ alS, const float* __restrict__ alD,
                             float* __restrict__ m) {
  int e = blockIdx.x * blockDim.x + threadIdx.x;
  if (e >= Etot) return;
  int is = gat_src_of(src, E, e);
  int id = gat_dst_of(dst, E, e);
#pragma unroll
  for (int h = 0; h < H; ++h) {
    float s  = alS[is * H + h] + alD[id * H + h];
    float lr = s > 0.f ? s : 0.2f * s;
    atomic_max_f32(&m[id * H + h], lr);
  }
}

// ---------------------------------------------------------------------------
// segment sum of exp(score - max) into dn[dst,head]
// ---------------------------------------------------------------------------
template <int H>
__global__ void gat_edge_den(const int* __restrict__ src, const int* __restrict__ dst,
                             int E, int Etot,
                             const float* __restrict__ alS, const float* __restrict__ alD,
                             const float* __restrict__ m, float* __restrict__ dn) {
  int e = blockIdx.x * blockDim.x + threadIdx.x;
  if (e >= Etot) return;
  int is = gat_src_of(src, E, e);
  int id = gat_dst_of(dst, E, e);
#pragma unroll
  for (int h = 0; h < H; ++h) {
    float s  = alS[is * H + h] + alD[id * H + h];
    float lr = s > 0.f ? s : 0.2f * s;
    atomic_add_f32(&dn[id * H + h], expf(lr - m[id * H + h]));
  }
}

// ---------------------------------------------------------------------------
// attention-weighted aggregation: out[dst] += alpha * h[src]
// one thread per (edge, 4-channel chunk); alpha recomputed in registers
// ---------------------------------------------------------------------------
template <int H, int C>
__global__ void gat_edge_agg(const int* __restrict__ src, const int* __restrict__ dst,
                             int E, int Etot,
                             const float* __restrict__ alS, const float* __restrict__ alD,
                             const float* __restrict__ m, const float* __restrict__ dn,
                             const float* __restrict__ hfeat, float* __restrict__ out) {
  constexpr int F   = H * C;
  constexpr int TPE = F / 4;
  long long t     = (long long)blockIdx.x * blockDim.x + threadIdx.x;
  long long total = (long long)Etot * TPE;
  if (t >= total) return;
  int e  = (int)(t / TPE);
  int q  = (int)(t - (long long)e * TPE);
  int ch = q * 4;
  int hd = ch / C;
  int is = gat_src_of(src, E, e);
  int id = gat_dst_of(dst, E, e);

  float s     = alS[is * H + hd] + alD[id * H + hd];
  float lr    = s > 0.f ? s : 0.2f * s;
  float alpha = expf(lr - m[id * H + hd]) / (dn[id * H + hd] + 1e-16f);

  const float4 hv = *reinterpret_cast<const float4*>(hfeat + (size_t)is * F + ch);
  float* o = out + (size_t)id * F + ch;
  atomic_add_f32(o + 0, hv.x * alpha);
  atomic_add_f32(o + 1, hv.y * alpha);
  atomic_add_f32(o + 2, hv.z * alpha);
  atomic_add_f32(o + 3, hv.w * alpha);
}

// ---------------------------------------------------------------------------
// utility / epilogue kernels
// ---------------------------------------------------------------------------
__global__ void gat_fill(float* __restrict__ p, long long n, float v) {
  long long i = (long long)blockIdx.x * blockDim.x + threadIdx.x;
  if (i < n) p[i] = v;
}

__global__ void gat_bias_elu(float* __restrict__ p, const float* __restrict__ b,
                             long long n, int F) {
  long long i = (long long)blockIdx.x * blockDim.x + threadIdx.x;
  if (i >= n) return;
  float v = p[i] + b[(int)(i % F)];
  p[i] = v > 0.f ? v : expm1f(v);
}

__global__ void gat_logsoftmax16(const float* __restrict__ agg, const float* __restrict__ b,
                                 float* __restrict__ out, int Nn) {
  int n = blockIdx.x * blockDim.x + threadIdx.x;
  if (n >= Nn) return;
  float v[16];
  float mx = -INFINITY;
#pragma unroll
  for (int c = 0; c < 16; ++c) {
    v[c] = agg[(size_t)n * 16 + c] + b[c];
    mx = fmaxf(mx, v[c]);
  }
  float s = 0.f;
#pragma unroll
  for (int c = 0; c < 16; ++c) s += expf(v[c] - mx);
  float ls = mx + logf(s);
#pragma unroll
  for (int c = 0; c < 16; ++c) out[(size_t)n * 16 + c] = v[c] - ls;
}

// ---------------------------------------------------------------------------
// host launcher
// ---------------------------------------------------------------------------
static inline int cdiv(long long a, long long b) { return (int)((a + b - 1) / b); }

extern "C" void kernel_launch(void* const* d_in, const int* in_sizes, int n_in,
                              void* d_out, int out_size, void* d_ws, size_t ws_size,
                              hipStream_t stream) {
  const float* x   = (const float*)d_in[0];
  const int*   ei  = (const int*)d_in[1];
  const float* W1  = (const float*)d_in[2];
  const float* aS1 = (const float*)d_in[3];
  const float* aD1 = (const float*)d_in[4];
  const float* b1  = (const float*)d_in[5];
  const float* W2  = (const float*)d_in[6];
  const float* aS2 = (const float*)d_in[7];
  const float* aD2 = (const float*)d_in[8];
  const float* b2  = (const float*)d_in[9];
  const float* W3  = (const float*)d_in[10];
  const float* aS3 = (const float*)d_in[11];
  const float* aD3 = (const float*)d_in[12];
  const float* b3  = (const float*)d_in[13];

  const int Nn   = in_sizes[0] / 128;     // 50000
  const int E    = in_sizes[1] / 2;       // 800000
  const int Etot = E + Nn;                // + self loops
  const int* src = ei;
  const int* dst = ei + E;

  char* wsb = (char*)d_ws;
  const size_t szBig = (size_t)Nn * 256 * sizeof(float);
  float* bufA = (float*)wsb;                         // projected features h
  float* bufB = (float*)(wsb + szBig);               // aggregated output / next input
  float* alS  = (float*)(wsb + 2 * szBig);
  float* alD  = alS + (size_t)Nn * 4;
  float* mb   = alD + (size_t)Nn * 4;
  float* db   = mb + (size_t)Nn * 4;
  float* Wt   = db + (size_t)Nn * 4;                 // transposed weights (<=128KB)

  const float NEG_INF = -INFINITY;
  const int mBlocks = cdiv(Nn, 16);

  // ======================= layer 1: 128 -> 4 x 64 =======================
  gat_transpose<<<cdiv(128 * 256, 256), 256, 0, stream>>>(W1, Wt, 128, 256);
  gat_gemm_wmma<128><<<mBlocks, 256, 0, stream>>>(x, Wt, bufA, Nn, 256);
  gat_node_logits<4, 64><<<cdiv((long long)Nn * 4, 256), 256, 0, stream>>>(
      bufA, aS1, aD1, alS, alD, Nn);
  gat_fill<<<cdiv((long long)Nn * 4, 256), 256, 0, stream>>>(mb, (long long)Nn * 4, NEG_INF);
  gat_fill<<<cdiv((long long)Nn * 4, 256), 256, 0, stream>>>(db, (long long)Nn * 4, 0.f);
  gat_fill<<<cdiv((long long)Nn * 256, 256), 256, 0, stream>>>(bufB, (long long)Nn * 256, 0.f);
  gat_edge_max<4><<<cdiv(Etot, 256), 256, 0, stream>>>(src, dst, E, Etot, alS, alD, mb);
  gat_edge_den<4><<<cdiv(Etot, 256), 256, 0, stream>>>(src, dst, E, Etot, alS, alD, mb, db);
  gat_edge_agg<4, 64><<<cdiv((long long)Etot * 64, 256), 256, 0, stream>>>(
      src, dst, E, Etot, alS, alD, mb, db, bufA, bufB);
  gat_bias_elu<<<cdiv((long long)Nn * 256, 256), 256, 0, stream>>>(
      bufB, b1, (long long)Nn * 256, 256);

  // ======================= layer 2: 256 -> 2 x 64 =======================
  gat_transpose<<<cdiv(256 * 128, 256), 256, 0, stream>>>(W2, Wt, 256, 128);
  gat_gemm_wmma<256><<<mBlocks, 256, 0, stream>>>(bufB, Wt, bufA, Nn, 128);
  gat_node_logits<2, 64><<<cdiv((long long)Nn * 2, 256), 256, 0, stream>>>(
      bufA, aS2, aD2, alS, alD, Nn);
  gat_fill<<<cdiv((long long)Nn * 2, 256), 256, 0, stream>>>(mb, (long long)Nn * 2, NEG_INF);
  gat_fill<<<cdiv((long long)Nn * 2, 256), 256, 0, stream>>>(db, (long long)Nn * 2, 0.f);
  gat_fill<<<cdiv((long long)Nn * 128, 256), 256, 0, stream>>>(bufB, (long long)Nn * 128, 0.f);
  gat_edge_max<2><<<cdiv(Etot, 256), 256, 0, stream>>>(src, dst, E, Etot, alS, alD, mb);
  gat_edge_den<2><<<cdiv(Etot, 256), 256, 0, stream>>>(src, dst, E, Etot, alS, alD, mb, db);
  gat_edge_agg<2, 64><<<cdiv((long long)Etot * 32, 256), 256, 0, stream>>>(
      src, dst, E, Etot, alS, alD, mb, db, bufA, bufB);
  gat_bias_elu<<<cdiv((long long)Nn * 128, 256), 256, 0, stream>>>(
      bufB, b2, (long long)Nn * 128, 128);

  // ======================= layer 3: 128 -> 1 x 16 =======================
  gat_transpose<<<cdiv(128 * 16, 256), 256, 0, stream>>>(W3, Wt, 128, 16);
  gat_gemm_wmma<128><<<mBlocks, 32, 0, stream>>>(bufB, Wt, bufA, Nn, 16);
  gat_node_logits<1, 16><<<cdiv(Nn, 256), 256, 0, stream>>>(bufA, aS3, aD3, alS, alD, Nn);
  gat_fill<<<cdiv(Nn, 256), 256, 0, stream>>>(mb, (long long)Nn, NEG_INF);
  gat_fill<<<cdiv(Nn, 256), 256, 0, stream>>>(db, (long long)Nn, 0.f);
  gat_fill<<<cdiv((long long)Nn * 16, 256), 256, 0, stream>>>(bufB, (long long)Nn * 16, 0.f);
  gat_edge_max<1><<<cdiv(Etot, 256), 256, 0, stream>>>(src, dst, E, Etot, alS, alD, mb);
  gat_edge_den<1><<<cdiv(Etot, 256), 256, 0, stream>>>(src, dst, E, Etot, alS, alD, mb, db);
  gat_edge_agg<1, 16><<<cdiv((long long)Etot * 4, 256), 256, 0, stream>>>(
      src, dst, E, Etot, alS, alD, mb, db, bufA, bufB);
  gat_logsoftmax16<<<cdiv(Nn, 256), 256, 0, stream>>>(bufB, b3, (float*)d_out, Nn);
}